// VoiceStitcher_43490838839623
// MI455X (gfx1250) — compile-verified
//
#include <hip/hip_runtime.h>
#include <hip/hip_bf16.h>
#include <cstdint>

// ---------------- problem constants (from reference) ----------------
#define BB 16
#define W 128
#define K 32
#define CC 9
#define NTR (W * K)        // 4096 tracks per batch
#define NEGV (-1.0e30f)
#define MAX_DF_ 0.05f
#define MAX_DPHI_ 0.5f
#define MAX_DA_ 0.5f
#define MIN_LEN 2

typedef __attribute__((ext_vector_type(2))) float v2f;
typedef __attribute__((ext_vector_type(8))) float v8f;

__device__ __forceinline__ float wrapf(float x) {
  const float PI_F = 3.14159265358979323846f;
  const float TWO_PI_F = 6.28318530717958647692f;
  float r = fmodf(x + PI_F, TWO_PI_F);   // python-mod semantics:
  r = (r < 0.0f) ? r + TWO_PI_F : r;     // result in [0, 2pi)
  return r - PI_F;
}

// =====================================================================
// Kernel 1: Viterbi DP. One wave per batch, lane = token k (wave32!).
// Cross-token values fetched with __shfl (warpSize==32 on gfx1250).
// =====================================================================
__global__ __launch_bounds__(32) void dp_kernel(const float* __restrict__ tok,
                                                float* __restrict__ totals,
                                                int* __restrict__ prev) {
  const int b = blockIdx.x;
  const int k = threadIdx.x;  // 0..31
  const float* tb = tok + (size_t)b * W * K * CC;

  float snr0 = tb[(0 * K + k) * CC + 0];
  bool pos0 = snr0 > 0.0f;
  float best = pos0 ? snr0 : NEGV;
  totals[(size_t)b * NTR + k] = best;   // where(pos0, snr0, NEG)
  prev[(size_t)b * NTR + k] = -1;

  // this lane's step-(w-1) values
  float p_fe = tb[(0 * K + k) * CC + 4];
  float p_pe = tb[(0 * K + k) * CC + 8];
  float p_ae = tb[(0 * K + k) * CC + 6];
  int flags = pos0 ? 3 : 0;  // bit0 = pos[w-1], bit1 = ext (DP carry)

  for (int w = 1; w < W; ++w) {
    const float* tw = tb + (size_t)(w * K + k) * CC;
    int wn = (w + 1 < W) ? (w + 1) : w;
    __builtin_prefetch(tb + (size_t)(wn * K + k) * CC, 0, 1);  // global_prefetch_b8

    float sn = tw[0];
    float fs = tw[3];
    float an = tw[5];
    float ps = tw[7];
    bool posc = sn > 0.0f;

    float bf = -INFINITY;
    int kpbest = 0;
    for (int kp = 0; kp < K; ++kp) {
      float fe = __shfl(p_fe, kp);
      float pe = __shfl(p_pe, kp);
      float ae = __shfl(p_ae, kp);
      float bkp = __shfl(best, kp);
      int fl = __shfl(flags, kp);

      float fm = (fe + fs) * 0.5f;
      bool ok_f = (fm <= 0.0f) ||
                  (fabsf(fe - fs) / ((fm > 0.0f) ? fm : 1.0f) <= MAX_DF_);
      bool ok_p = fabsf(wrapf(ps - pe)) <= MAX_DPHI_;
      float am = fmaxf(ae, an);
      bool ok_a = (am <= 0.0f) ||
                  (fabsf(ae - an) / ((am > 0.0f) ? am : 1.0f) <= MAX_DA_);
      bool E = (fl & 1) && posc && ok_f && ok_p && ok_a;
      float cand = (E && (fl & 2)) ? bkp : NEGV;
      if (cand > bf) { bf = cand; kpbest = kp; }  // strict > == first argmax
    }
    bool reached = bf > NEGV * 0.5f;
    float te = bf + sn;
    float tot = reached ? te : (posc ? sn : NEGV);
    int prv = reached ? kpbest : -1;
    totals[(size_t)b * NTR + (size_t)w * K + k] = tot;
    prev[(size_t)b * NTR + (size_t)w * K + k] = prv;

    best = reached ? te : NEGV;
    p_fe = tw[4]; p_pe = tw[8]; p_ae = tw[6];
    flags = (posc ? 1 : 0) | (reached ? 2 : 0);
  }
}

// =====================================================================
// Kernel 2: path backtrace table P[n][j] (int8 token id or -1).
// Sequential over w, LDS ping-pong of the K x W slice.
// =====================================================================
__global__ __launch_bounds__(256) void path_kernel(const float* __restrict__ tok,
                                                   const int* __restrict__ prev,
                                                   int8_t* __restrict__ P) {
  __shared__ int8_t rows[2][K][W];  // 8 KB
  const int b = blockIdx.x;
  const int tid = threadIdx.x;
  const float* tb = tok + (size_t)b * W * K * CC;
  int8_t* Pb = P + (size_t)b * NTR * W;
  const int* prevb = prev + (size_t)b * NTR;

  for (int i = tid; i < K * W; i += 256) {
    int k = i / W, j = i % W;
    bool p0 = tb[(size_t)k * CC] > 0.0f;
    int8_t v = (p0 && j == 0) ? (int8_t)k : (int8_t)-1;
    rows[0][k][j] = v;
    Pb[(size_t)k * W + j] = v;
  }
  __syncthreads();

  for (int w = 1; w < W; ++w) {
    int cur = w & 1, prv = cur ^ 1;
    for (int i = tid; i < K * W; i += 256) {
      int k = i / W, j = i % W;
      int pv = prevb[w * K + k];
      int8_t v = (pv >= 0) ? rows[prv][pv][j] : (int8_t)-1;
      if (j == w) v = (tb[(size_t)(w * K + k) * CC] > 0.0f) ? (int8_t)k : (int8_t)-1;
      rows[cur][k][j] = v;
      Pb[(size_t)(w * K + k) * W + j] = v;
    }
    __syncthreads();
  }
}

// =====================================================================
// Kernel 3: stable descending sort (bitonic over 64-bit composite keys)
// + sequential greedy conflict selection (wave32 ballot per candidate).
// =====================================================================
__global__ __launch_bounds__(256) void select_kernel(const float* __restrict__ tok,
                                                     const float* __restrict__ totals,
                                                     const int8_t* __restrict__ P,
                                                     uint8_t* __restrict__ selected) {
  __shared__ unsigned long long keys[NTR];  // 32 KB
  __shared__ uint8_t used[NTR];             // 4 KB
  const int b = blockIdx.x;
  const int tid = threadIdx.x;
  const float* tb = tok + (size_t)b * W * K * CC;
  const float* totb = totals + (size_t)b * NTR;

  for (int n = tid; n < NTR; n += 256) {
    float snr = tb[(size_t)n * CC];
    float tf = (snr > 0.0f) ? totb[n] : -INFINITY;
    unsigned u = __float_as_uint(tf);
    u ^= (u >> 31) ? 0xFFFFFFFFu : 0x80000000u;  // orderable mapping
    // secondary key: smaller n first on ties (matches stable argsort)
    keys[n] = ((unsigned long long)u << 32) | (unsigned)(NTR - 1 - n);
  }
  for (unsigned size = 2; size <= NTR; size <<= 1) {
    for (unsigned stride = size >> 1; stride; stride >>= 1) {
      __syncthreads();
      for (unsigned i = tid; i < NTR / 2; i += 256) {
        unsigned lo = (i << 1) - (i & (stride - 1));
        unsigned hi = lo + stride;
        bool desc = ((lo & size) == 0);
        unsigned long long a = keys[lo], c = keys[hi];
        if ((a < c) == desc) { keys[lo] = c; keys[hi] = a; }
      }
    }
  }
  __syncthreads();

  if (tid < 32) {  // one wave does the sequential greedy pass
    const int lane = tid;
    for (int i = lane; i < NTR; i += 32) used[i] = 0;
    const int8_t* Pb = P + (size_t)b * NTR * W;
    uint8_t* selb = selected + (size_t)b * NTR;
    const unsigned NEGINF_KEY = 0x007FFFFFu;  // orderable(-inf)

    for (int r = 0; r < NTR; ++r) {
      unsigned long long key = keys[r];
      int n = NTR - 1 - (int)(unsigned)(key & 0xFFFFFFFFu);
      bool feasible = ((unsigned)(key >> 32)) != NEGINF_KEY;
      const int8_t* row = Pb + (size_t)n * W;
      if (r + 4 < NTR) {
        int n2 = NTR - 1 - (int)(unsigned)(keys[r + 4] & 0xFFFFFFFFu);
        __builtin_prefetch(Pb + (size_t)n2 * W + lane * 4, 0, 1);
      }
      bool conflict = false;
      int rv[4];
#pragma unroll
      for (int q = 0; q < 4; ++q) {
        int j = lane + q * 32;
        int v = row[j];
        rv[q] = v;
        conflict |= (v >= 0) && (used[j * K + v] != 0);
      }
      bool ok = feasible && (__ballot(conflict) == 0ull);
      if (ok) {
#pragma unroll
        for (int q = 0; q < 4; ++q) {
          int j = lane + q * 32;
          if (rv[q] >= 0) used[j * K + rv[q]] = 1;
        }
      }
      if (lane == 0) selb[n] = ok ? 1 : 0;
      __builtin_amdgcn_wave_barrier();  // order cross-lane LDS updates
    }
  }
}

// =====================================================================
// Kernel Z: zero-fill the 135 MB output at full store bandwidth.
// =====================================================================
__global__ void zero_kernel(float4* __restrict__ out, size_t n4) {
  size_t i = (size_t)blockIdx.x * blockDim.x + threadIdx.x;
  size_t stride = (size_t)gridDim.x * blockDim.x;
  float4 z = make_float4(0.f, 0.f, 0.f, 0.f);
  for (; i < n4; i += stride) out[i] = z;
}

// =====================================================================
// Kernel 4: anchors. One wave handles 16 tracks. The phase prefix sum
// cum = T(strict-lower-ones, 128x128) x D(128x16) runs on the tensor
// pipe via V_WMMA_F32_16X16X4_F32 (f32 in/out => exact).
// =====================================================================
__global__ __launch_bounds__(32) void anchor_kernel(const float* __restrict__ tok,
                                                    const int8_t* __restrict__ P,
                                                    const uint8_t* __restrict__ selected,
                                                    float* __restrict__ out) {
  const int b = blockIdx.x >> 8;    // 256 groups of 16 tracks per batch
  const int g = blockIdx.x & 255;
  const int n0 = g * 16;
  const int lane = threadIdx.x;
  const int t = lane & 15, h = lane >> 4;
  const int w_end = n0 >> 5;        // uniform: 16 | 32 so n/K constant in group

  const float* tb = tok + (size_t)b * W * K * CC;
  __shared__ int8_t rowbuf[16][W];  // 2 KB
  __shared__ float dmat[W][16];     // 8 KB
  __shared__ float cum[W][16];      // 8 KB (exclusive prefix sums)
  __shared__ int Ls[16];
  __shared__ uint8_t selS[16];

  {  // stage 16 contiguous path rows (2 KB) into LDS
    const uint32_t* src = (const uint32_t*)(P + ((size_t)b * NTR + n0) * W);
    uint32_t* dst = (uint32_t*)&rowbuf[0][0];
    for (int i = lane; i < 16 * W / 4; i += 32) dst[i] = src[i];
  }
  if (lane < 16) selS[lane] = selected[(size_t)b * NTR + n0 + lane];
  __syncthreads();
  if (lane < 16) {
    int L = 0;
    for (int j = 0; j < W; ++j) L += (rowbuf[lane][j] >= 0) ? 1 : 0;
    Ls[lane] = L;
  }
  __syncthreads();

  bool mysel = (lane < 16) && selS[lane] && (Ls[lane] >= MIN_LEN);
  if (__ballot(mysel) == 0ull) return;  // uniform exit; zero_kernel covered us

  {  // per-track phase deltas d[j], zero outside [ws, w_end]
    int L = Ls[t];
    bool sel = selS[t] && (L >= MIN_LEN);
    int ws = w_end - L + 1;
    for (int j = h * 64; j < h * 64 + 64; ++j) {
      float d = 0.0f;
      if (sel && j >= ws && j <= w_end) {
        int rc = max((int)rowbuf[t][j], 0);
        float psj = tb[(size_t)(j * K + rc) * CC + 7];
        float pej = tb[(size_t)(j * K + rc) * CC + 8];
        if (j == ws) {
          d = pej - psj;
        } else {
          int rcm = max((int)rowbuf[t][j - 1], 0);
          float pem = tb[(size_t)((j - 1) * K + rcm) * CC + 8];
          d = wrapf(psj - pem) + pej - psj;
        }
      }
      dmat[j][t] = d;
    }
  }
  __syncthreads();

#if __has_builtin(__builtin_amdgcn_wmma_f32_16x16x4_f32)
  {  // cum = T x D on the tensor pipe, 16x16x4 f32 tiles (EXEC all ones here)
    const int m = lane & 15;          // A row / C col / B col
    const int kh = (lane >> 4) * 2;   // K lanes: 0..15 -> K{0,1}, 16..31 -> K{2,3}
    for (int pi = 0; pi < 8; ++pi) {
      v8f acc = {};
      const int kkmax = pi * 4 + 4;   // only chunks with j < pi*16+16 contribute
      for (int kk = 0; kk < kkmax; ++kk) {
        int j0 = kk * 4 + kh;
        int p = pi * 16 + m;
        v2f a, bv;
        a.x = (j0 < p) ? 1.0f : 0.0f;        // strict lower-triangular T
        a.y = (j0 + 1 < p) ? 1.0f : 0.0f;
        bv.x = dmat[j0][m];
        bv.y = dmat[j0 + 1][m];
        acc = __builtin_amdgcn_wmma_f32_16x16x4_f32(false, a, false, bv,
                                                    (short)0, acc, false, false);
      }
#pragma unroll
      for (int r = 0; r < 8; ++r) {   // C layout: vgpr r -> M=r (lanes<16) / r+8
        int p = pi * 16 + ((lane < 16) ? r : (r + 8));
        cum[p][m] = acc[r];
      }
    }
  }
#else
  if (lane < 16) {
    float run = 0.0f;
    for (int j = 0; j < W; ++j) { cum[j][lane] = run; run += dmat[j][lane]; }
  }
#endif
  __syncthreads();

  {  // emit anchors for selected tracks only (rest already zeroed)
    int L = Ls[t];
    bool sel = selS[t] && (L >= MIN_LEN);
    if (sel) {
      int ws = w_end - L + 1;
      int n = n0 + t;
      int rc0 = max((int)rowbuf[t][ws], 0);
      float ps0 = tb[(size_t)(ws * K + rc0) * CC + 7];
      float4* ob = (float4*)(out + ((size_t)b * NTR + n) * (size_t)(W + 1) * 4);
      for (int q = h; q <= L; q += 2) {
        int p = ws + q;
        float aA, aT, aF, aPHI;
        if (q == 0) {  // start anchor
          const float* tp = tb + (size_t)(ws * K + rc0) * CC;
          aT = tp[1]; aF = tp[3]; aA = tp[5];
          aPHI = ps0;
        } else if (p == w_end + 1) {  // end anchor
          int rc = max((int)rowbuf[t][w_end], 0);
          const float* tp = tb + (size_t)(w_end * K + rc) * CC;
          aT = tp[2]; aF = tp[4]; aA = tp[6];
          float cfull = (p == W) ? (cum[W - 1][t] + dmat[W - 1][t]) : cum[p][t];
          aPHI = ps0 + cfull;
        } else {  // interior midpoint
          int rcm = max((int)rowbuf[t][p - 1], 0);
          int rc = max((int)rowbuf[t][p], 0);
          const float* tm = tb + (size_t)((p - 1) * K + rcm) * CC;
          const float* tp = tb + (size_t)(p * K + rc) * CC;
          aT = (tm[2] + tp[1]) * 0.5f;
          aF = (tm[4] + tp[3]) * 0.5f;
          aA = (tm[6] + tp[5]) * 0.5f;
          aPHI = ps0 + cum[p][t];
        }
        ob[q] = make_float4(aA, aT, aPHI, aF);  // anch = [A, t, phi, f]
      }
    }
  }
}

// =====================================================================
extern "C" void kernel_launch(void* const* d_in, const int* in_sizes, int n_in,
                              void* d_out, int out_size, void* d_ws, size_t ws_size,
                              hipStream_t stream) {
  (void)in_sizes; (void)n_in; (void)ws_size;
  const float* tok = (const float*)d_in[0];
  float* out = (float*)d_out;

  char* ws = (char*)d_ws;                                  // layout (~8.6 MB):
  float* totals = (float*)(ws);                            // B*N f32
  int* prev = (int*)(ws + (size_t)BB * NTR * 4);           // B*N i32
  int8_t* P = (int8_t*)(ws + (size_t)BB * NTR * 8);        // B*N*W i8
  uint8_t* selected =
      (uint8_t*)(ws + (size_t)BB * NTR * 8 + (size_t)BB * NTR * W);  // B*N u8

  dp_kernel<<<BB, 32, 0, stream>>>(tok, totals, prev);
  path_kernel<<<BB, 256, 0, stream>>>(tok, prev, P);
  select_kernel<<<BB, 256, 0, stream>>>(tok, totals, P, selected);

  size_t n4 = (size_t)out_size / 4;  // out_size = 16*4096*129*4, % 4 == 0
  zero_kernel<<<4096, 256, 0, stream>>>((float4*)out, n4);
  anchor_kernel<<<BB * 256, 32, 0, stream>>>(tok, P, selected, out);
}